// DivrocLoss_14714557956152
// MI455X (gfx1250) — compile-verified
//
#include <hip/hip_runtime.h>

#define GR   128
#define VOX  (GR * GR * GR)   // 2^21 voxels

typedef __attribute__((ext_vector_type(2))) float v2f;
typedef __attribute__((ext_vector_type(8))) float v8f;

// ---------------------------------------------------------------------------
// Trilinear scatter-add of one point with weight w into the 128^3 grid.
// Adjoint of grid_sample(align_corners=False, zeros padding):
//   pix = ((p + 1) * 128 - 1) / 2 = 64*p + 63.5
// ---------------------------------------------------------------------------
__device__ __forceinline__ void splat_point(float* __restrict__ grid,
                                            float px, float py, float pz,
                                            float w) {
    float x = px * 64.0f + 63.5f;
    float y = py * 64.0f + 63.5f;
    float z = pz * 64.0f + 63.5f;
    float x0 = floorf(x), y0 = floorf(y), z0 = floorf(z);
    float fx = x - x0, fy = y - y0, fz = z - z0;
    int xi = (int)x0, yi = (int)y0, zi = (int)z0;
    float wxs[2] = {1.0f - fx, fx};
    float wys[2] = {1.0f - fy, fy};
    float wzs[2] = {1.0f - fz, fz};
#pragma unroll
    for (int dz = 0; dz < 2; ++dz) {
        int zc = zi + dz;
        if (zc < 0 || zc >= GR) continue;
        float wz = w * wzs[dz];
#pragma unroll
        for (int dy = 0; dy < 2; ++dy) {
            int yc = yi + dy;
            if (yc < 0 || yc >= GR) continue;
            float wzy = wz * wys[dy];
#pragma unroll
            for (int dx = 0; dx < 2; ++dx) {
                int xc = xi + dx;
                if (xc < 0 || xc >= GR) continue;
                // hardware global_atomic_add_f32, no return (STOREcnt only)
                unsafeAtomicAdd(&grid[(zc * GR + yc) * GR + xc], wzy * wxs[dx]);
            }
        }
    }
}

// One thread per point: splat pred point with +1 and gt point with -1 into one
// diff grid (splatting is linear; Huber only needs pred_rast - gt_rast).
__global__ void __launch_bounds__(256)
splat_diff_kernel(const float* __restrict__ pred,
                  const float* __restrict__ gt,
                  const float* __restrict__ coords,
                  float* __restrict__ grid, int n) {
    int i = blockIdx.x * blockDim.x + threadIdx.x;
    if (i >= n) return;
    float cx = coords[3 * i + 0];
    float cy = coords[3 * i + 1];
    float cz = coords[3 * i + 2];
    splat_point(grid, cx + pred[3 * i + 0], cy + pred[3 * i + 1],
                cz + pred[3 * i + 2], 1.0f);
    splat_point(grid, cx + gt[3 * i + 0], cy + gt[3 * i + 1],
                cz + gt[3 * i + 2], -1.0f);
}

// ---------------------------------------------------------------------------
// Huber(diff, delta=1) summed over the grid, accumulated with
// V_WMMA_F32_16X16X4_F32:  A = ones(16x4)  =>  D[m][n] = colsum_n(B) + C[m][n].
// Straight-line: 2 x b128 loads per lane up-front, 4 WMMAs on 2 independent
// accumulators. 1024 blocks x 256 threads x 8 floats == 128^3 exactly, so no
// guards -> EXEC is all-1s for every WMMA (hardware requirement).
// ---------------------------------------------------------------------------
#define RED_BLOCKS  1024
#define RED_THREADS 256
#define RED_LANES   (RED_BLOCKS * RED_THREADS)   // 262144; x8 floats = VOX

__device__ __forceinline__ float huber1(float d) {
    float ad = fabsf(d);
    return (ad <= 1.0f) ? 0.5f * d * d : (ad - 0.5f);
}

// wave32 XOR-butterfly add via ds_swizzle (group-of-32 mode, and_mask=0x1F)
#define SWZ_ADD(v, offs)                                                     \
    (v) += __int_as_float(                                                   \
        __builtin_amdgcn_ds_swizzle(__float_as_int(v), (offs)))

__global__ void __launch_bounds__(RED_THREADS)
huber_wmma_reduce_kernel(const float* __restrict__ grid,
                         float* __restrict__ out) {
    const int tid = blockIdx.x * RED_THREADS + threadIdx.x;
    const float4* g4 = (const float4*)grid;

    // issue both 128-bit loads before any math
    float4 p0 = g4[tid];
    float4 p1 = g4[tid + RED_LANES];

    v2f a;                  // ones A-matrix (16x4 f32 = 2 VGPRs/lane)
    a.x = 1.0f; a.y = 1.0f;
    v8f c0 = {}, c1 = {};   // two independent f32 accumulators
    v2f b;

    b.x = huber1(p0.x); b.y = huber1(p0.y);
    c0 = __builtin_amdgcn_wmma_f32_16x16x4_f32(false, a, false, b, (short)0,
                                               c0, false, false);
    b.x = huber1(p1.x); b.y = huber1(p1.y);
    c1 = __builtin_amdgcn_wmma_f32_16x16x4_f32(false, a, false, b, (short)0,
                                               c1, false, false);
    b.x = huber1(p0.z); b.y = huber1(p0.w);
    c0 = __builtin_amdgcn_wmma_f32_16x16x4_f32(false, a, false, b, (short)0,
                                               c0, false, false);
    b.x = huber1(p1.z); b.y = huber1(p1.w);
    c1 = __builtin_amdgcn_wmma_f32_16x16x4_f32(false, a, false, b, (short)0,
                                               c1, false, false);

    // c[0] = one column sum per lane; each column appears twice across the 32
    // lanes (lane n and lane n+16) -> full-wave XOR reduce gives 2x wave total.
    float s = c0[0] + c1[0];
    SWZ_ADD(s, 0x401F);   // xor 16
    SWZ_ADD(s, 0x201F);   // xor 8
    SWZ_ADD(s, 0x101F);   // xor 4
    SWZ_ADD(s, 0x081F);   // xor 2
    SWZ_ADD(s, 0x041F);   // xor 1
    s *= 0.5f;

    __shared__ float lds[RED_THREADS / 32];
    const int lane = threadIdx.x & 31;
    const int wave = threadIdx.x >> 5;
    if (lane == 0) lds[wave] = s;
    __syncthreads();
    if (threadIdx.x == 0) {
        float t = 0.0f;
#pragma unroll
        for (int w = 0; w < RED_THREADS / 32; ++w) t += lds[w];
        unsafeAtomicAdd(out, t);
    }
}

extern "C" void kernel_launch(void* const* d_in, const int* in_sizes, int n_in,
                              void* d_out, int out_size, void* d_ws,
                              size_t ws_size, hipStream_t stream) {
    const float* pred   = (const float*)d_in[0];  // registration_pred [1,N,3]
    const float* gt     = (const float*)d_in[1];  // registration_gt   [1,N,3]
    const float* coords = (const float*)d_in[2];  // coords            [1,N,3]
    float* out  = (float*)d_out;
    float* grid = (float*)d_ws;                   // 128^3 f32 diff grid (8 MB)
    const int n = in_sizes[2] / 3;

    hipMemsetAsync(grid, 0, (size_t)VOX * sizeof(float), stream);
    hipMemsetAsync(out, 0, sizeof(float), stream);

    const int threads = 256;
    const int blocks  = (n + threads - 1) / threads;
    splat_diff_kernel<<<blocks, threads, 0, stream>>>(pred, gt, coords, grid, n);

    huber_wmma_reduce_kernel<<<RED_BLOCKS, RED_THREADS, 0, stream>>>(grid, out);
}